// GATv2NodeFeatures_88527865905194
// MI455X (gfx1250) — compile-verified
//
#include <hip/hip_runtime.h>
#include <math.h>

typedef __attribute__((ext_vector_type(2))) float v2f;
typedef __attribute__((ext_vector_type(4))) float f32x4;
typedef __attribute__((ext_vector_type(8))) float v8f;

#define NEG_SLOPE 0.2f

constexpr int B_  = 4;
constexpr int V_  = 512;
constexpr int H_  = 128;
constexpr int BV_ = B_ * V_;   // 2048

__device__ __forceinline__ float wave_reduce_sum(float v) {
#pragma unroll
  for (int off = 16; off > 0; off >>= 1)
    v += __shfl_xor(v, off, 32);
  return v;
}

__device__ __forceinline__ float wave_reduce_max(float v) {
#pragma unroll
  for (int off = 16; off > 0; off >>= 1)
    v = fmaxf(v, __shfl_xor(v, off, 32));
  return v;
}

// ---------------------------------------------------------------------------
// Kernel 1: edge_mask[b,i,j] = mean(edge_gate[b,i,j,:]) over H=128.
// Pure bandwidth (512 MB streamed once). One wave per row: each lane does one
// 128-bit non-temporal load (32 lanes x 4 floats = 128), then wave reduce.
// ---------------------------------------------------------------------------
__global__ void edge_mean_kernel(const float* __restrict__ eg,
                                 float* __restrict__ mask, int nrows) {
  const int wave   = (blockIdx.x * blockDim.x + threadIdx.x) >> 5;
  const int lane   = threadIdx.x & 31;
  const int nwaves = (gridDim.x * blockDim.x) >> 5;
  for (int row = wave; row < nrows; row += nwaves) {
    const f32x4* p = (const f32x4*)(eg + (size_t)row * H_) + lane;
    f32x4 v = __builtin_nontemporal_load(p);
    float s = (v.x + v.y) + (v.z + v.w);
    s = wave_reduce_sum(s);
    if (lane == 0) mask[row] = s * (1.0f / (float)H_);
  }
}

// ---------------------------------------------------------------------------
// Kernel 2/6: D[M x 128] = A[M x 128] @ W[128 x 128]^T + bias
// One wave per 16x16 tile, f32 WMMA, K-loop of 32 x v_wmma_f32_16x16x4_f32.
// A-frag: lane l -> row (l&15), K = 2*(l>>4)+{0,1}. B-frag mirrors with
// N = (l&15); here B[k][n] = W[n][k], so both frags load along W rows.
// ---------------------------------------------------------------------------
__global__ void gemm_nt_bias_wmma(const float* __restrict__ A,
                                  const float* __restrict__ W,
                                  const float* __restrict__ bias,
                                  float* __restrict__ D) {
  const int wave  = (blockIdx.x * blockDim.x + threadIdx.x) >> 5;
  const int lane  = threadIdx.x & 31;
  const int tileM = wave >> 3;   // M/16 tiles (grid-sized)
  const int tileN = wave & 7;    // 128/16 = 8 N tiles
  const int half  = lane >> 4;
  const int idx16 = lane & 15;

  const float* arow = A + (size_t)(tileM * 16 + idx16) * H_;
  const float* wrow = W + (size_t)(tileN * 16 + idx16) * H_;

  v8f acc = {};
  for (int k = 0; k < H_; k += 4) {
    const int kk = k + half * 2;
    v2f av = { arow[kk], arow[kk + 1] };
    v2f bv = { wrow[kk], wrow[kk + 1] };
    acc = __builtin_amdgcn_wmma_f32_16x16x4_f32(false, av, false, bv,
                                                (short)0, acc, false, false);
  }

  const int colg = tileN * 16 + idx16;
  const float bc = bias[colg];
#pragma unroll
  for (int r = 0; r < 8; ++r) {
    const int rowg = tileM * 16 + r + 8 * half;
    D[(size_t)rowg * H_ + colg] = acc[r] + bc;
  }
}

// ---------------------------------------------------------------------------
// Kernel 3: s_i[v] = dot(h[v,:], a_i_flat), s_j[v] = dot(h[v,:], a_j_flat)
// a has shape (1, NH=8, 2*HD=32); a_i = a[0,:, :16], a_j = a[0,:,16:].
// Flat h index j -> head n=j/16, k=j%16; a_i idx = n*32+k, a_j = n*32+16+k.
// One wave per row, 4 floats per lane.
// ---------------------------------------------------------------------------
__global__ void attn_scores_kernel(const float* __restrict__ h,
                                   const float* __restrict__ a,
                                   float* __restrict__ s_i,
                                   float* __restrict__ s_j) {
  const int wave = (blockIdx.x * blockDim.x + threadIdx.x) >> 5;
  const int lane = threadIdx.x & 31;
  if (wave >= BV_) return;

  const f32x4 hv = *(const f32x4*)(h + (size_t)wave * H_ + lane * 4);
  const int n     = lane >> 2;         // (lane*4)/16
  const int kbase = (lane & 3) * 4;    // k offset within head

  float di = 0.f, dj = 0.f;
#pragma unroll
  for (int t = 0; t < 4; ++t) {
    const float hvt = hv[t];
    di += hvt * a[n * 32 + kbase + t];
    dj += hvt * a[n * 32 + 16 + kbase + t];
  }
  di = wave_reduce_sum(di);
  dj = wave_reduce_sum(dj);
  if (lane == 0) { s_i[wave] = di; s_j[wave] = dj; }
}

// ---------------------------------------------------------------------------
// Kernel 4: per (b,i) row: scores_j = leaky_relu(s_i+s_j)*mask, softmax over j.
// One wave per row of V=512: 16 values per lane, shuffle max/sum reductions.
// ---------------------------------------------------------------------------
__global__ void softmax_kernel(const float* __restrict__ s_i,
                               const float* __restrict__ s_j,
                               const float* __restrict__ mask,
                               float* __restrict__ wts) {
  const int wave = (blockIdx.x * blockDim.x + threadIdx.x) >> 5;
  const int lane = threadIdx.x & 31;
  if (wave >= BV_) return;

  const int b = wave >> 9;  // / V_
  const float si = s_i[wave];
  const float* sjb  = s_j + (size_t)b * V_;
  const float* mrow = mask + (size_t)wave * V_;

  float vals[16];
  float mx = -INFINITY;
#pragma unroll
  for (int t = 0; t < 16; ++t) {
    const int j = lane + t * 32;
    float sc = si + sjb[j];
    sc = (sc > 0.f) ? sc : NEG_SLOPE * sc;
    sc *= mrow[j];
    vals[t] = sc;
    mx = fmaxf(mx, sc);
  }
  mx = wave_reduce_max(mx);

  float sum = 0.f;
#pragma unroll
  for (int t = 0; t < 16; ++t) {
    vals[t] = __expf(vals[t] - mx);
    sum += vals[t];
  }
  sum = wave_reduce_sum(sum);
  const float inv = 1.0f / sum;

  float* wrow = wts + (size_t)wave * V_;
#pragma unroll
  for (int t = 0; t < 16; ++t) wrow[lane + t * 32] = vals[t] * inv;
}

// ---------------------------------------------------------------------------
// Kernel 5: attended[b] = weights[b] (512x512) @ h[b] (512x128), f32 WMMA.
// One wave per 16x16 tile; K-loop of 128 x v_wmma_f32_16x16x4_f32.
// ---------------------------------------------------------------------------
__global__ void attend_wmma(const float* __restrict__ wts,
                            const float* __restrict__ h,
                            float* __restrict__ att) {
  const int wave  = (blockIdx.x * blockDim.x + threadIdx.x) >> 5;
  const int lane  = threadIdx.x & 31;
  const int b     = wave >> 8;     // 256 tiles per batch
  const int tile  = wave & 255;
  const int tileM = tile >> 3;     // 0..31
  const int tileN = tile & 7;      // 0..7
  const int half  = lane >> 4;
  const int idx16 = lane & 15;

  const float* arow = wts + ((size_t)b * V_ + tileM * 16 + idx16) * V_;
  const float* bmat = h + (size_t)b * V_ * H_;   // row-major [k][n]
  const int coln = tileN * 16 + idx16;

  v8f acc = {};
  for (int k = 0; k < V_; k += 4) {
    const int kk = k + half * 2;
    v2f av = { arow[kk], arow[kk + 1] };
    v2f bv = { bmat[(size_t)kk * H_ + coln],
               bmat[(size_t)(kk + 1) * H_ + coln] };
    acc = __builtin_amdgcn_wmma_f32_16x16x4_f32(false, av, false, bv,
                                                (short)0, acc, false, false);
  }

#pragma unroll
  for (int r = 0; r < 8; ++r) {
    const int rowg = tileM * 16 + r + 8 * half;
    att[((size_t)b * V_ + rowg) * H_ + coln] = acc[r];
  }
}

// ---------------------------------------------------------------------------
extern "C" void kernel_launch(void* const* d_in, const int* in_sizes, int n_in,
                              void* d_out, int out_size, void* d_ws, size_t ws_size,
                              hipStream_t stream) {
  (void)in_sizes; (void)n_in; (void)out_size; (void)ws_size;

  const float* x    = (const float*)d_in[0];   // (4,512,128)
  const float* eg   = (const float*)d_in[1];   // (4,512,512,128)
  const float* W_w  = (const float*)d_in[2];   // (128,128)
  const float* W_b  = (const float*)d_in[3];   // (128,)
  const float* a    = (const float*)d_in[4];   // (1,8,32)
  const float* Wo_w = (const float*)d_in[5];   // (128,128)
  const float* Wo_b = (const float*)d_in[6];   // (128,)
  float* out = (float*)d_out;                  // (4,512,128)

  // Workspace layout (floats), ~10.5 MB total.
  float* ws   = (float*)d_ws;
  float* mask = ws;                         // 4*512*512   = 1048576
  float* h    = mask + (size_t)BV_ * V_;    // 2048*128    =  262144
  float* si   = h + (size_t)BV_ * H_;       // 2048
  float* sj   = si + BV_;                   // 2048
  float* wts  = sj + BV_;                   // 1048576
  float* att  = wts + (size_t)BV_ * V_;     // 262144

  // 1) edge mask: 1,048,576 rows of 128 floats (the 512 MB stream).
  edge_mean_kernel<<<4096, 256, 0, stream>>>(eg, mask, BV_ * V_);

  // 2) h = x @ W_w^T + W_b : 2048x128, 1024 tile-waves.
  gemm_nt_bias_wmma<<<128, 256, 0, stream>>>(x, W_w, W_b, h);

  // 3) s_i, s_j : one wave per row (2048 waves).
  attn_scores_kernel<<<256, 256, 0, stream>>>(h, a, si, sj);

  // 4) softmax weights : one wave per (b,i) row (2048 waves).
  softmax_kernel<<<256, 256, 0, stream>>>(si, sj, mask, wts);

  // 5) attended = weights @ h (batched) : 1024 tile-waves.
  attend_wmma<<<128, 256, 0, stream>>>(wts, h, att);

  // 6) out = attended @ Wo_w^T + Wo_b : 1024 tile-waves.
  gemm_nt_bias_wmma<<<128, 256, 0, stream>>>(att, Wo_w, Wo_b, out);
}